// EnsembleGRUModule_63213328662935
// MI455X (gfx1250) — compile-verified
//
#include <hip/hip_runtime.h>

typedef float f32x2 __attribute__((ext_vector_type(2)));
typedef float f32x8 __attribute__((ext_vector_type(8)));

#define W_T     64
#define E_N     16
#define BI_N    2560
#define F_N     8
#define CHUNK   128              // bi sequences per workgroup
#define BLOCK   128              // 4 waves (wave32)
#define NWAVES  (BLOCK / 32)
#define TILES   (W_T * (CHUNK / 16))   // 16-row N-tiles per workgroup = 512

// fast device helpers: raw v_exp_f32 / v_rcp_f32 (~1 ulp), no division ladder
__device__ __forceinline__ float fast_sigmoid(float x) {
    return __builtin_amdgcn_rcpf(1.f + __expf(-x));
}
__device__ __forceinline__ float fast_tanh(float x) {
    // tanh(x) = 1 - 2/(e^{2x}+1); exp(+inf)->inf, rcp(inf)->0 => +1; exp->0 => -1
    return 1.f - 2.f * __builtin_amdgcn_rcpf(__expf(2.f * x) + 1.f);
}

__global__ __launch_bounds__(BLOCK)
void gru_fused_wmma(const float* __restrict__ x,     // (W,E,BI,F)
                    const float* __restrict__ state, // (1,E,BI,1)
                    const float* __restrict__ wl,    // (E,16,8)
                    const float* __restrict__ bl,    // (E,16)
                    const float* __restrict__ wih,   // (E,3,16)
                    const float* __restrict__ whh,   // (E,3,1)
                    const float* __restrict__ bih,   // (E,3)
                    const float* __restrict__ bhh,   // (E,3)
                    float* __restrict__ out)         // (W,E,BI,1)
{
    __shared__ float gi_lds[W_T * CHUNK * 3];  // 96 KB: gi[w][bi_local][gate]
    __shared__ float sWc[3][F_N];              // folded weights (Wih @ Wl), 3x8
    __shared__ float sbc[3];                   // folded bias
    __shared__ float swhh[3];
    __shared__ float sbhh[3];

    const int e   = blockIdx.y;
    const int bi0 = blockIdx.x * CHUNK;
    const int t   = threadIdx.x;

    // ---- fold the two einsum weights into one 3x8 matrix per ensemble member ----
    if (t < 24) {
        int g = t >> 3, f = t & 7;
        float s = 0.f;
        for (int o = 0; o < 16; ++o)
            s += wih[(e * 3 + g) * 16 + o] * wl[(e * 16 + o) * 8 + f];
        sWc[g][f] = s;
    }
    if (t < 3) {
        float s = bih[e * 3 + t];
        for (int o = 0; o < 16; ++o)
            s += wih[(e * 3 + t) * 16 + o] * bl[e * 16 + o];
        sbc[t]  = s;
        swhh[t] = whh[e * 3 + t];
        sbhh[t] = bhh[e * 3 + t];
    }
    __syncthreads();

    const int wave = t >> 5;
    const int lane = t & 31;
    const int m    = lane & 15;   // row-in-halfwave
    const int hv   = lane >> 4;   // 0 -> K{0,1}, 1 -> K{2,3}

    // A-matrix (16x4 f32): gates on M, rows 3..15 zero.
    // Lane L<16: (K0,K1) of row L ; lane L+16: (K2,K3) of row L.
    f32x2 a_lo, a_hi;
    a_lo.x = (m < 3) ? sWc[m][hv * 2 + 0] : 0.f;
    a_lo.y = (m < 3) ? sWc[m][hv * 2 + 1] : 0.f;
    a_hi.x = (m < 3) ? sWc[m][4 + hv * 2 + 0] : 0.f;
    a_hi.y = (m < 3) ? sWc[m][4 + hv * 2 + 1] : 0.f;

    // C bias: C[m][n] = bc[m]; lanes 16..31 carry M=8..15 -> zero.
    f32x8 cb;
    #pragma unroll
    for (int r = 0; r < 8; ++r)
        cb[r] = (hv == 0 && r < 3) ? sbc[r] : 0.f;

    // ---- phase 1: gi = Wc @ x + bc via V_WMMA_F32_16X16X4_F32, staged in LDS ----
    for (int tile = wave; tile < TILES; tile += NWAVES) {
        int w   = tile >> 3;            // CHUNK/16 == 8 tiles per timestep
        int bib = (tile & 7) << 4;
        // B-matrix (4x16 f32): column N = input row; lane L<16 holds (K0,K1),
        // lane L+16 holds (K2,K3) of row L  -> contiguous float2 loads.
        const float* xr = x + (((size_t)(w * E_N + e) * BI_N) + bi0 + bib + m) * F_N
                            + hv * 2;
        f32x2 b_lo = *(const f32x2*)(xr);       // K 0..3 slice
        f32x2 b_hi = *(const f32x2*)(xr + 4);   // K 4..7 slice

        f32x8 d = __builtin_amdgcn_wmma_f32_16x16x4_f32(
            false, a_lo, false, b_lo, (short)0, cb, false, false);
        d = __builtin_amdgcn_wmma_f32_16x16x4_f32(
            false, a_hi, false, b_hi, (short)0, d, false, false);

        // D VGPR r, lanes 0..15 = gate r of row (bib+lane). Stride-3 words: no bank conflicts.
        if (lane < 16) {
            int base = (w * CHUNK + bib + lane) * 3;
            gi_lds[base + 0] = d[0];
            gi_lds[base + 1] = d[1];
            gi_lds[base + 2] = d[2];
        }
    }
    __syncthreads();

    // ---- phase 2: sequential GRU recurrence, one lane per sequence ----
    float h = state[e * BI_N + bi0 + t];
    const float wh0 = swhh[0], wh1 = swhh[1], wh2 = swhh[2];
    const float bh0 = sbhh[0], bh1 = sbhh[1], bh2 = sbhh[2];

    for (int w = 0; w < W_T; ++w) {
        int base  = (w * CHUNK + t) * 3;
        float g0  = gi_lds[base + 0] + wh0 * h + bh0;
        float g1  = gi_lds[base + 1] + wh1 * h + bh1;
        float g2  = gi_lds[base + 2];
        float gh2 = wh2 * h + bh2;
        float r   = fast_sigmoid(g0);
        float z   = fast_sigmoid(g1);
        float nn  = fast_tanh(g2 + r * gh2);
        h = (1.f - z) * nn + z * h;
        out[((size_t)(w * E_N + e)) * BI_N + bi0 + t] = h;
    }
}

extern "C" void kernel_launch(void* const* d_in, const int* in_sizes, int n_in,
                              void* d_out, int out_size, void* d_ws, size_t ws_size,
                              hipStream_t stream) {
    const float* x     = (const float*)d_in[0];
    const float* state = (const float*)d_in[1];
    const float* wl    = (const float*)d_in[2];
    const float* bl    = (const float*)d_in[3];
    const float* wih   = (const float*)d_in[4];
    const float* whh   = (const float*)d_in[5];
    const float* bih   = (const float*)d_in[6];
    const float* bhh   = (const float*)d_in[7];
    float* out = (float*)d_out;

    dim3 grid(BI_N / CHUNK, E_N);  // 20 x 16 = 320 workgroups
    gru_fused_wmma<<<grid, BLOCK, 0, stream>>>(x, state, wl, bl, wih, whh, bih, bhh, out);
}